// Decoder_31705448579434
// MI455X (gfx1250) — compile-verified
//
#include <hip/hip_runtime.h>
#include <hip/hip_bf16.h>
#include <math.h>

#define XS   256
#define XS2  (XS * XS)          // 65536
#define BB   32
#define KS   5

typedef __attribute__((ext_vector_type(2))) float v2f;
typedef __attribute__((ext_vector_type(8))) float v8f;

// ---------------------------------------------------------------------------
// V_WMMA_F32_16X16X4_F32 wrapper: D = A(16x4) * B(4x16) + C(16x16), all f32.
// NEG[1:0] must be 0 for f32 per ISA 7.12, so negation happens in VALU.
// ---------------------------------------------------------------------------
static __device__ inline v8f wmma4(v2f a, v2f b, v8f c) {
    return __builtin_amdgcn_wmma_f32_16x16x4_f32(
        false, a, false, b, (short)0, c, false, false);
}

// ---------------------------------------------------------------------------
// DFT twiddle tables.  W[u,y] = exp(-2*pi*i*u*y/256)   (forward)
//                      V[u,y] = exp(+2*pi*i*u*y/256)/256 (inverse, per axis)
// Both are symmetric: W[u,y] == W[y,u].
// ---------------------------------------------------------------------------
__global__ void init_dft_tables(float* __restrict__ Wre, float* __restrict__ Wim,
                                float* __restrict__ Vre, float* __restrict__ Vim) {
    int i = blockIdx.x;          // 0..255
    int j = threadIdx.x;         // 0..255
    int t = (i * j) & (XS - 1);  // (i*j) mod 256 keeps the angle small
    float ang = -6.283185307179586f * (float)t / (float)XS;
    float s, c;
    __sincosf(ang, &s, &c);
    int idx = i * XS + j;
    Wre[idx] = c;
    Wim[idx] = s;
    Vre[idx] = c * (1.0f / (float)XS);
    Vim[idx] = -s * (1.0f / (float)XS);
}

// ---------------------------------------------------------------------------
// Per-batch: rot6d -> R, rotate/project points, bilinear scatter into a full
// 256KB LDS image (ds_add_f32), then 5x5 zero-padded conv, write to global.
// One workgroup (1024 threads = 32 waves) per batch image.
// ---------------------------------------------------------------------------
__global__ __launch_bounds__(1024)
void scatter_conv_kernel(const float* __restrict__ alignment,
                         const float* __restrict__ shifts,
                         const float* __restrict__ coords,
                         const float* __restrict__ values,
                         const float* __restrict__ gk,
                         float* __restrict__ img_out,
                         int npts) {
    extern __shared__ float simg[];        // XS2 floats = 256 KB
    __shared__ float sk[KS * KS];

    const int b    = blockIdx.x;
    const int tid  = threadIdx.x;
    const int nthr = blockDim.x;

    for (int i = tid; i < XS2; i += nthr) simg[i] = 0.0f;
    if (tid < KS * KS) sk[tid] = gk[tid];

    // rot6d -> rotation matrix (redundant per thread; cheap)
    float a1x = alignment[b * 6 + 0], a1y = alignment[b * 6 + 1], a1z = alignment[b * 6 + 2];
    float a2x = alignment[b * 6 + 3], a2y = alignment[b * 6 + 4], a2z = alignment[b * 6 + 5];
    float n1 = sqrtf(a1x * a1x + a1y * a1y + a1z * a1z) + 1e-8f;
    float b1x = a1x / n1, b1y = a1y / n1, b1z = a1z / n1;
    float d = b1x * a2x + b1y * a2y + b1z * a2z;
    float p2x = a2x - d * b1x, p2y = a2y - d * b1y, p2z = a2z - d * b1z;
    float n2 = sqrtf(p2x * p2x + p2y * p2y + p2z * p2z) + 1e-8f;
    float b2x = p2x / n2, b2y = p2y / n2, b2z = p2z / n2;
    float sx = shifts[b * 2 + 0] + (float)(XS / 2);
    float sy = shifts[b * 2 + 1] + (float)(XS / 2);

    __syncthreads();

    for (int n = tid; n < npts; n += nthr) {
        float cx = coords[n * 3 + 0], cy = coords[n * 3 + 1], cz = coords[n * 3 + 2];
        float px = b1x * cx + b1y * cy + b1z * cz + sx;
        float py = b2x * cx + b2y * cy + b2z * cz + sy;
        float x0 = floorf(px), y0 = floorf(py);
        float fx = px - x0, fy = py - y0;
        int x0i = (int)x0, y0i = (int)y0;
        float v = values[n];

        float w00 = (1.0f - fx) * (1.0f - fy);
        float w01 = fx * (1.0f - fy);
        float w10 = (1.0f - fx) * fy;
        float w11 = fx * fy;

        int xc0 = min(max(x0i, 0), XS - 1);
        int xc1 = min(max(x0i + 1, 0), XS - 1);
        int yc0 = min(max(y0i, 0), XS - 1);
        int yc1 = min(max(y0i + 1, 0), XS - 1);

        atomicAdd(&simg[yc0 * XS + xc0], v * w00);
        atomicAdd(&simg[yc0 * XS + xc1], v * w01);
        atomicAdd(&simg[yc1 * XS + xc0], v * w10);
        atomicAdd(&simg[yc1 * XS + xc1], v * w11);
    }

    __syncthreads();

    for (int p = tid; p < XS2; p += nthr) {
        int y = p >> 8, x = p & (XS - 1);
        float acc = 0.0f;
        #pragma unroll
        for (int dy = 0; dy < KS; ++dy) {
            int yy = y + dy - 2;
            if (yy < 0 || yy >= XS) continue;
            #pragma unroll
            for (int dx = 0; dx < KS; ++dx) {
                int xx = x + dx - 2;
                if (xx < 0 || xx >= XS) continue;
                acc += sk[dy * KS + dx] * simg[yy * XS + xx];
            }
        }
        img_out[b * XS2 + p] = acc;
    }
}

// ---------------------------------------------------------------------------
// Complex 256x256 GEMM on WMMA f32 16x16x4.  C = A * B (complex).
//   HASBIM : B is complex
//   HASCIM : emit Im(C)
//   BSYM   : B is symmetric (DFT twiddle table) -> fetch B fragments from
//            rows instead of columns, turning 2 strided b32 loads into one
//            contiguous b64 load per fragment register pair.
// One wave computes a 16x32 slab (two N-tiles); A fragments and the negated
// Aim fragment are reused across both WMMA chains.
// Grid: (8 N-slabs, 4 M-tile groups, batch), block 128 (4 waves).
// ---------------------------------------------------------------------------
template <bool HASBIM, bool HASCIM, bool BSYM>
__global__ __launch_bounds__(128)
void cgemm256_wmma(const float* __restrict__ Are, const float* __restrict__ Aim, int aBatched,
                   const float* __restrict__ Bre, const float* __restrict__ Bim, int bBatched,
                   float* __restrict__ Cre, float* __restrict__ Cim) {
    const int lane  = threadIdx.x & 31;
    const int wave  = threadIdx.x >> 5;
    const int mBase = (blockIdx.y * 4 + wave) * 16;
    const int nBase = blockIdx.x * 32;
    const int b     = blockIdx.z;

    const int aOff = aBatched ? b * XS2 : 0;
    const int bOff = bBatched ? b * XS2 : 0;
    const int cOff = b * XS2;

    // A fragment (16x4): lanes 0-15 -> M=lane, K=k+{0,1}; lanes 16-31 -> K=k+{2,3}
    const int am    = mBase + (lane & 15);
    const int akoff = (lane >> 4) << 1;
    // B fragment (4x16): vgpr v, lanes 0-15 -> K=k+v; lanes 16-31 -> K=k+v+2
    const int bn    = nBase + (lane & 15);
    const int bkoff = (lane >> 4) << 1;

    const float* arow = Are + aOff + am * XS + akoff;
    const float* aimr = Aim + aOff + am * XS + akoff;

    // Column-walk pointers (generic B) or row-walk pointers (symmetric B).
    const float* bre0 = BSYM ? (Bre + bOff + bn * XS + bkoff)
                             : (Bre + bOff + bkoff * XS + bn);
    const float* bre1 = BSYM ? (bre0 + 16 * XS) : (bre0 + 16);
    const float* bim0 = HASBIM ? (BSYM ? (Bim + bOff + bn * XS + bkoff)
                                       : (Bim + bOff + bkoff * XS + bn))
                               : nullptr;
    const float* bim1 = HASBIM ? (BSYM ? (bim0 + 16 * XS) : (bim0 + 16)) : nullptr;

    v8f cre0 = {}, cre1 = {};
    v8f cim0 = {}, cim1 = {};

    #pragma unroll 4
    for (int k = 0; k < XS; k += 4) {
        v2f ar, ai;
        ar.x = arow[k];     ar.y = arow[k + 1];
        ai.x = aimr[k];     ai.y = aimr[k + 1];

        v2f br0, br1;
        if (BSYM) {
            br0.x = bre0[k];        br0.y = bre0[k + 1];
            br1.x = bre1[k];        br1.y = bre1[k + 1];
        } else {
            br0.x = bre0[k * XS];   br0.y = bre0[(k + 1) * XS];
            br1.x = bre1[k * XS];   br1.y = bre1[(k + 1) * XS];
        }

        cre0 = wmma4(ar, br0, cre0);
        cre1 = wmma4(ar, br1, cre1);
        if (HASCIM) {
            cim0 = wmma4(ai, br0, cim0);
            cim1 = wmma4(ai, br1, cim1);
        }

        if (HASBIM) {
            v2f bi0, bi1;
            if (BSYM) {
                bi0.x = bim0[k];        bi0.y = bim0[k + 1];
                bi1.x = bim1[k];        bi1.y = bim1[k + 1];
            } else {
                bi0.x = bim0[k * XS];   bi0.y = bim0[(k + 1) * XS];
                bi1.x = bim1[k * XS];   bi1.y = bim1[(k + 1) * XS];
            }
            v2f nai = -ai;                        // f32 WMMA has no A/B neg modifier
            cre0 = wmma4(nai, bi0, cre0);         // Re += -Aim*Bim
            cre1 = wmma4(nai, bi1, cre1);
            if (HASCIM) {
                cim0 = wmma4(ar, bi0, cim0);      // Im += Are*Bim
                cim1 = wmma4(ar, bi1, cim1);
            }
        }
    }

    // C/D layout: vgpr i -> M = mBase + i + 8*(lane>=16), N = nBase + (lane&15)
    const int cn   = nBase + (lane & 15);
    const int mAdd = (lane >> 4) << 3;
    #pragma unroll
    for (int i = 0; i < 8; ++i) {
        int row = cOff + (mBase + i + mAdd) * XS + cn;
        Cre[row]      = cre0[i];
        Cre[row + 16] = cre1[i];
        if (HASCIM) {
            Cim[row]      = cim0[i];
            Cim[row + 16] = cim1[i];
        }
    }
}

// ---------------------------------------------------------------------------
// Spectrum * circularly-shifted CTF (fftshift == ifftshift for even size).
// ---------------------------------------------------------------------------
__global__ void ctf_mul_kernel(float* __restrict__ Yre, float* __restrict__ Yim,
                               const float* __restrict__ ctf) {
    int b = blockIdx.x;
    int u = blockIdx.y;
    int v = threadIdx.x;
    float c = ctf[b * XS2 + (((u + XS / 2) & (XS - 1)) * XS) + ((v + XS / 2) & (XS - 1))];
    int idx = b * XS2 + u * XS + v;
    Yre[idx] *= c;
    Yim[idx] *= c;
}

// ---------------------------------------------------------------------------
extern "C" void kernel_launch(void* const* d_in, const int* in_sizes, int n_in,
                              void* d_out, int out_size, void* d_ws, size_t ws_size,
                              hipStream_t stream) {
    const float* alignment = (const float*)d_in[0];
    const float* shifts    = (const float*)d_in[1];
    const float* coords    = (const float*)d_in[2];
    const float* values    = (const float*)d_in[3];
    const float* gk        = (const float*)d_in[4];
    const float* ctf       = (const float*)d_in[5];
    const int npts = in_sizes[3];

    float* ws = (float*)d_ws;
    float* img = ws;                       size_t o = (size_t)BB * XS2;   // conv'd images
    float* Wre = ws + o; o += XS2;
    float* Wim = ws + o; o += XS2;
    float* Vre = ws + o; o += XS2;
    float* Vim = ws + o; o += XS2;
    float* Xre = ws + o; o += (size_t)BB * XS2;
    float* Xim = ws + o; o += (size_t)BB * XS2;
    float* Yre = ws + o; o += (size_t)BB * XS2;
    float* Yim = ws + o; o += (size_t)BB * XS2;

    // 1) DFT tables
    init_dft_tables<<<dim3(XS), XS, 0, stream>>>(Wre, Wim, Vre, Vim);

    // 2) scatter + 5x5 conv, full image resident in LDS
    scatter_conv_kernel<<<dim3(BB), 1024, XS2 * sizeof(float), stream>>>(
        alignment, shifts, coords, values, gk, img, npts);

    dim3 ggrid(XS / 32, 4, BB);   // 8 N-slabs x (4 blocks * 4 waves = 16 M-tiles)

    // 3) X = W * img            (B real batched, emit complex)
    cgemm256_wmma<false, true, false><<<ggrid, 128, 0, stream>>>(Wre, Wim, 0, img, nullptr, 1, Xre, Xim);
    // 4) Y = X * W              (full complex, B symmetric twiddle -> row loads)
    cgemm256_wmma<true, true, true><<<ggrid, 128, 0, stream>>>(Xre, Xim, 1, Wre, Wim, 0, Yre, Yim);
    // 5) Y *= ifftshift(ctf)
    ctf_mul_kernel<<<dim3(BB, XS), XS, 0, stream>>>(Yre, Yim, ctf);
    // 6) X = V * Y              (full complex, B batched -> column loads)
    cgemm256_wmma<true, true, false><<<ggrid, 128, 0, stream>>>(Vre, Vim, 0, Yre, Yim, 1, Xre, Xim);
    // 7) out = Re(X * V)        (real-only output, B symmetric twiddle)
    cgemm256_wmma<true, false, true><<<ggrid, 128, 0, stream>>>(Xre, Xim, 1, Vre, Vim, 0, (float*)d_out, nullptr);
}